// Net_39960375722314
// MI455X (gfx1250) — compile-verified
//
#include <hip/hip_runtime.h>

typedef __attribute__((ext_vector_type(2))) float v2f;
typedef __attribute__((ext_vector_type(8))) float v8f;

// ---------------- degree / normalization ----------------

__global__ void k_deg_init(float* __restrict__ deg, int n) {
  int i = blockIdx.x * blockDim.x + threadIdx.x;
  if (i < n) deg[i] = 1.0f;  // self-loop contributes 1 to every node's degree
}

__global__ void k_deg_edges(const int* __restrict__ dst, float* __restrict__ deg, int e) {
  int i = blockIdx.x * blockDim.x + threadIdx.x;
  if (i < e) atomicAdd(&deg[dst[i]], 1.0f);
}

__global__ void k_rsqrt_inplace(float* __restrict__ deg, int n) {
  int i = blockIdx.x * blockDim.x + threadIdx.x;
  if (i < n) deg[i] = rsqrtf(deg[i]);  // deg >= 1 always (self-loop)
}

// ---------------- strip-mined WMMA f32 GEMM ----------------
// One wave computes a full 16-row x MOUT strip: the A fragment (16x4 f32) is
// loaded once per K-step and reused across all MOUT/16 column tiles, so each
// row of X is fetched exactly once. Optionally fuses the PREVIOUS layer's
// bias+ReLU into the A-load (ACT), removing a full N*MOUT read+write pass.
// Epilogue applies dinv scaling: hs = h*dinv[row], agg = h*dinv[row]^2 (self-loop).
//
// VGPR layouts (ISA 7.12.2):
//  A 16x4:  lane<16 holds row=lane, K=k+{0,1}; lane>=16 holds row=lane-16, K=k+{2,3}
//  B 4x16:  lane<16 holds col=lane, K=k+{0,1}; lane>=16 holds col=lane-16, K=k+{2,3}
//  C/D:     element v -> row = 8*(lane>=16) + v, col = lane&15

template <int K, int MOUT, bool ACT>
__global__ void k_gemm_strip(const float* __restrict__ X, const float* __restrict__ W,
                             const float* __restrict__ bias_in,  // prev layer bias (ACT only)
                             const float* __restrict__ dinv,
                             float* __restrict__ hs, float* __restrict__ agg, int nrows) {
  constexpr int T = MOUT / 16;  // column tiles per strip
  const int wave = (blockIdx.x * blockDim.x + threadIdx.x) >> 5;
  const int lane = threadIdx.x & 31;
  if (wave * 16 >= nrows) return;  // wave-uniform: EXEC all-1s for WMMA

  const int half = lane >> 4;
  const int l16  = lane & 15;
  const int rowA = wave * 16 + l16;

  v8f acc[T];
#pragma unroll
  for (int t = 0; t < T; ++t) acc[t] = (v8f){};

#pragma unroll 4
  for (int k = 0; k < K; k += 4) {
    const int ka = k + 2 * half;
    v2f a;
    a.x = X[(size_t)rowA * K + ka];
    a.y = X[(size_t)rowA * K + ka + 1];
    if (ACT) {  // fused bias+ReLU of previous layer's aggregation
      a.x = fmaxf(a.x + bias_in[ka], 0.0f);
      a.y = fmaxf(a.y + bias_in[ka + 1], 0.0f);
    }
#pragma unroll
    for (int t = 0; t < T; ++t) {
      const int colB = t * 16 + l16;
      v2f b;
      b.x = W[(size_t)ka * MOUT + colB];
      b.y = W[(size_t)(ka + 1) * MOUT + colB];
      acc[t] = __builtin_amdgcn_wmma_f32_16x16x4_f32(
          false, a, false, b, (short)0, acc[t], false, false);
    }
  }

#pragma unroll
  for (int v = 0; v < 8; ++v) {
    const int row = wave * 16 + v + 8 * half;
    const float dn = dinv[row];
#pragma unroll
    for (int t = 0; t < T; ++t) {
      const int col = t * 16 + l16;
      const float h = acc[t][v] * dn;             // hs = h * dinv[row]
      hs[(size_t)row * MOUT + col]  = h;
      agg[(size_t)row * MOUT + col] = h * dn;     // self-loop init: h * dinv^2
    }
  }
}

// ---------------- edge scatter-add: agg[dst] += hs[src] * dinv[dst] ----------------
// One wave per edge: the 512B/256B/128B feature row is gathered with wide
// vector loads (b128/b64), then one global_atomic_add_f32 per element.
// All per-layer agg buffers (<=51MB) are L2-resident (192MB L2).

template <int MOUT>
__global__ void k_scatter(const int* __restrict__ src, const int* __restrict__ dst,
                          const float* __restrict__ hs, const float* __restrict__ dinv,
                          float* __restrict__ agg, int e) {
  constexpr int VPT = MOUT / 32;  // floats per lane: 4, 4, 2, 1
  const long long g = (long long)blockIdx.x * blockDim.x + threadIdx.x;
  const int edge = (int)(g >> 5);
  const int lane = (int)(g & 31);
  if (edge >= e) return;
  const int s = src[edge];
  const int d = dst[edge];
  const float dn = dinv[d];
  const float* hrow = hs  + (size_t)s * MOUT + lane * VPT;
  float*       arow = agg + (size_t)d * MOUT + lane * VPT;
  if constexpr (VPT == 4) {
    const float4 v = *(const float4*)hrow;
    atomicAdd(arow + 0, v.x * dn);
    atomicAdd(arow + 1, v.y * dn);
    atomicAdd(arow + 2, v.z * dn);
    atomicAdd(arow + 3, v.w * dn);
  } else if constexpr (VPT == 2) {
    const float2 v = *(const float2*)hrow;
    atomicAdd(arow + 0, v.x * dn);
    atomicAdd(arow + 1, v.y * dn);
  } else {
    atomicAdd(arow, hrow[0] * dn);
  }
}

// ---------------- final bias (output layer only, no ReLU) ----------------

template <int MOUT>
__global__ void k_bias_out(const float* __restrict__ agg, const float* __restrict__ bias,
                           float* __restrict__ out, int n) {
  const size_t i = (size_t)blockIdx.x * blockDim.x + threadIdx.x;
  if (i >= (size_t)n * MOUT) return;
  out[i] = agg[i] + bias[(int)(i % MOUT)];
}

// ---------------- host-side layer driver ----------------

template <int K, int MOUT, bool ACT>
static void run_layer(const float* in, const float* W, const float* bias_in,
                      const float* dinv, const int* esrc, const int* edst,
                      float* hs, float* agg, int n, int e, hipStream_t stream) {
  const long long strips = (n + 15) / 16;
  const int gblocks = (int)((strips * 32 + 255) / 256);
  k_gemm_strip<K, MOUT, ACT><<<gblocks, 256, 0, stream>>>(in, W, bias_in, dinv, hs, agg, n);

  const int sblocks = (int)(((long long)e * 32 + 255) / 256);
  k_scatter<MOUT><<<sblocks, 256, 0, stream>>>(esrc, edst, hs, dinv, agg, e);
}

extern "C" void kernel_launch(void* const* d_in, const int* in_sizes, int n_in,
                              void* d_out, int out_size, void* d_ws, size_t ws_size,
                              hipStream_t stream) {
  const float* x  = (const float*)d_in[0];
  const int*   ei = (const int*)d_in[1];
  const float* W1 = (const float*)d_in[2];
  const float* b1 = (const float*)d_in[3];
  const float* W2 = (const float*)d_in[4];
  const float* b2 = (const float*)d_in[5];
  const float* W3 = (const float*)d_in[6];
  const float* b3 = (const float*)d_in[7];
  const float* W4 = (const float*)d_in[8];
  const float* b4 = (const float*)d_in[9];

  const int n = in_sizes[0] / 256;  // 100000
  const int e = in_sizes[1] / 2;    // 1600000
  const int* esrc = ei;             // edge_index[0]
  const int* edst = ei + e;         // edge_index[1]

  // workspace layout (floats): dinv[n] | H[n*128] | G1[n*128] | G2[n*128]
  float* dinv = (float*)d_ws;
  float* H  = dinv + n;
  float* G1 = H  + (size_t)n * 128;
  float* G2 = G1 + (size_t)n * 128;

  // ---- normalization: dinv = rsqrt(1 + in-degree) ----
  k_deg_init<<<(n + 255) / 256, 256, 0, stream>>>(dinv, n);
  k_deg_edges<<<(e + 255) / 256, 256, 0, stream>>>(edst, dinv, e);
  k_rsqrt_inplace<<<(n + 255) / 256, 256, 0, stream>>>(dinv, n);

  // ---- 4 GCN layers; bias+ReLU of layer l fused into GEMM of layer l+1 ----
  run_layer<256, 128, false>(x,  W1, nullptr, dinv, esrc, edst, H, G1, n, e, stream);
  run_layer<128, 128, true >(G1, W2, b1,      dinv, esrc, edst, H, G2, n, e, stream);
  run_layer<128, 64,  true >(G2, W3, b2,      dinv, esrc, edst, H, G1, n, e, stream);
  run_layer<64,  32,  true >(G1, W4, b3,      dinv, esrc, edst, H, G2, n, e, stream);

  // ---- output: z = agg4 + b4 (no ReLU) ----
  const long long tot = (long long)n * 32;
  k_bias_out<32><<<(int)((tot + 255) / 256), 256, 0, stream>>>(G2, b4, (float*)d_out, n);
}